// Resnet50RoIHead_4715874091193
// MI455X (gfx1250) — compile-verified
//
#include <hip/hip_runtime.h>
#include <hip/hip_bf16.h>
#include <math.h>

typedef __attribute__((ext_vector_type(2))) float v2f;
typedef __attribute__((ext_vector_type(8))) float v8f;

// Problem constants (fixed by setup_inputs)
constexpr int Nimg = 2;
constexpr int Cc   = 1024;
constexpr int Hf   = 38;
constexpr int Wf   = 38;
constexpr int Rr   = 128;
constexpr int RS   = 14;      // ROI_SIZE
constexpr int NFC  = 2048;    // fc7 width
constexpr int NC   = 84;      // n_classes*4
constexpr int NS   = 21;      // n_classes
constexpr int NT   = NC + NS; // 105 combined head columns

// ---------------------------------------------------------------------------
// Kernel 1: fused RoIPool(max) + mean over 14x14 bins -> pooled (128,1024)
// One thread per (roi, channel). x is NCHW; per-thread inner w-loop walks
// contiguous floats, and the whole 11.8MB tensor is L2-resident.
// ---------------------------------------------------------------------------
__global__ void roipool_mean_kernel(const float* __restrict__ x,
                                    const float* __restrict__ rois,
                                    const int*   __restrict__ roi_idx,
                                    const int*   __restrict__ img_size,
                                    float*       __restrict__ pooled) {
  const int roi = blockIdx.x;             // 0..127
  const int c   = blockIdx.y * blockDim.x + threadIdx.x; // 0..1023

  const float img_h = (float)img_size[0];
  const float img_w = (float)img_size[1];
  const float sx = (float)Wf / img_w;
  const float sy = (float)Hf / img_h;

  const float rx1 = rois[roi * 4 + 0] * sx;
  const float ry1 = rois[roi * 4 + 1] * sy;
  const float rx2 = rois[roi * 4 + 2] * sx;
  const float ry2 = rois[roi * 4 + 3] * sy;

  // round-to-nearest-even to match jnp.round
  const int x1 = (int)rintf(rx1);
  const int y1 = (int)rintf(ry1);
  const int x2 = (int)rintf(rx2);
  const int y2 = (int)rintf(ry2);

  const float roi_w = fmaxf((float)(x2 - x1 + 1), 1.0f);
  const float roi_h = fmaxf((float)(y2 - y1 + 1), 1.0f);
  const float bin_h = roi_h / (float)RS;
  const float bin_w = roi_w / (float)RS;

  const int n = roi_idx[roi];
  const float* __restrict__ plane = x + ((size_t)n * Cc + c) * (size_t)(Hf * Wf);

  float sum = 0.0f;
  for (int ph = 0; ph < RS; ++ph) {
    int hs = min(max(y1 + (int)floorf((float)ph * bin_h), 0), Hf);
    int he = min(max(y1 + (int)ceilf((float)(ph + 1) * bin_h), 0), Hf);
    for (int pw = 0; pw < RS; ++pw) {
      int ws = min(max(x1 + (int)floorf((float)pw * bin_w), 0), Wf);
      int we = min(max(x1 + (int)ceilf((float)(pw + 1) * bin_w), 0), Wf);
      const bool empty = (he <= hs) || (we <= ws);
      float m = -INFINITY;
      for (int h = hs; h < he; ++h) {
        const float* rowp = plane + h * Wf;
        for (int w = ws; w < we; ++w) m = fmaxf(m, rowp[w]);
      }
      sum += empty ? 0.0f : m;
    }
  }
  pooled[(size_t)roi * Cc + c] = sum * (1.0f / (float)(RS * RS));
}

// ---------------------------------------------------------------------------
// Kernel 2: fc7 = pooled(128x1024) @ W_feat(1024x2048) + b_feat, fp32 WMMA.
// One wave computes a 16x32 slab (two 16x16 N-tiles) to reuse the A fragment.
// K stepped by 4 with V_WMMA_F32_16X16X4_F32.
// ---------------------------------------------------------------------------
__global__ void gemm_fc7_kernel(const float* __restrict__ A,    // 128x1024
                                const float* __restrict__ B,    // 1024x2048
                                const float* __restrict__ bias, // 2048
                                float*       __restrict__ Cout) // 128x2048
{
  constexpr int K = Cc;   // 1024
  constexpr int N = NFC;  // 2048
  const int wave = (blockIdx.x * blockDim.x + threadIdx.x) >> 5;
  const int lane = threadIdx.x & 31;
  const int tm  = wave & 7;       // 8 M-tiles
  const int tnp = wave >> 3;      // 64 N-pairs
  const int lo = lane & 15;
  const int hi = lane >> 4;

  const int arow_i = tm * 16 + lo;
  const int col0 = tnp * 32 + lo;
  const int col1 = col0 + 16;

  const float* __restrict__ arow = A + (size_t)arow_i * K;

  v8f acc0 = {};
  v8f acc1 = {};
  for (int k = 0; k < K; k += 4) {
    const int kk = k + 2 * hi;
    v2f a;
    a.x = arow[kk];
    a.y = arow[kk + 1];
    v2f b0, b1;
    b0.x = B[(size_t)kk * N + col0];
    b0.y = B[(size_t)(kk + 1) * N + col0];
    b1.x = B[(size_t)kk * N + col1];
    b1.y = B[(size_t)(kk + 1) * N + col1];
    acc0 = __builtin_amdgcn_wmma_f32_16x16x4_f32(false, a, false, b0,
                                                 (short)0, acc0, false, false);
    acc1 = __builtin_amdgcn_wmma_f32_16x16x4_f32(false, a, false, b1,
                                                 (short)0, acc1, false, false);
  }

  const float bb0 = bias[col0];
  const float bb1 = bias[col1];
#pragma unroll
  for (int g = 0; g < 8; ++g) {
    const int r = tm * 16 + g + 8 * hi;
    Cout[(size_t)r * N + col0] = acc0[g] + bb0;
    Cout[(size_t)r * N + col1] = acc1[g] + bb1;
  }
}

// ---------------------------------------------------------------------------
// Kernel 3: heads. out = fc7(128x2048) @ [W_cls_loc | W_score] + bias.
// Combined N=105, padded to 7 tiles of 16 (zero B columns beyond 105).
// d_out layout: 128*84 cls_locs then 128*21 scores (flat == reshape views).
// ---------------------------------------------------------------------------
__global__ void gemm_heads_kernel(const float* __restrict__ fc7, // 128x2048
                                  const float* __restrict__ Wc,  // 2048x84
                                  const float* __restrict__ bc,  // 84
                                  const float* __restrict__ Ws,  // 2048x21
                                  const float* __restrict__ bs,  // 21
                                  float*       __restrict__ out)
{
  constexpr int K = NFC; // 2048
  const int wave = (blockIdx.x * blockDim.x + threadIdx.x) >> 5;
  const int lane = threadIdx.x & 31;
  const int tm = wave & 7;   // 8 M-tiles (waves laid out tm-fastest)
  const int tn = wave >> 3;  // 7 N-tiles
  const int lo = lane & 15;
  const int hi = lane >> 4;

  const int row = tm * 16 + lo;
  const int col = tn * 16 + lo;   // combined-head column (0..111)

  const float* __restrict__ arow = fc7 + (size_t)row * K;

  v8f acc = {};
  for (int k = 0; k < K; k += 4) {
    const int kk = k + 2 * hi;
    v2f a;
    a.x = arow[kk];
    a.y = arow[kk + 1];
    float b0, b1;
    if (col < NC) {
      b0 = Wc[(size_t)kk * NC + col];
      b1 = Wc[(size_t)(kk + 1) * NC + col];
    } else if (col < NT) {
      b0 = Ws[(size_t)kk * NS + (col - NC)];
      b1 = Ws[(size_t)(kk + 1) * NS + (col - NC)];
    } else {
      b0 = 0.0f;
      b1 = 0.0f;
    }
    v2f b;
    b.x = b0;
    b.y = b1;
    acc = __builtin_amdgcn_wmma_f32_16x16x4_f32(false, a, false, b,
                                                (short)0, acc, false, false);
  }

#pragma unroll
  for (int g = 0; g < 8; ++g) {
    const int r = tm * 16 + g + 8 * hi;
    if (col < NC) {
      out[(size_t)r * NC + col] = acc[g] + bc[col];
    } else if (col < NT) {
      out[(size_t)Rr * NC + (size_t)r * NS + (col - NC)] = acc[g] + bs[col - NC];
    }
  }
}

// ---------------------------------------------------------------------------
extern "C" void kernel_launch(void* const* d_in, const int* in_sizes, int n_in,
                              void* d_out, int out_size, void* d_ws, size_t ws_size,
                              hipStream_t stream) {
  const float* x        = (const float*)d_in[0];
  const float* rois     = (const float*)d_in[1];
  const int*   roi_idx  = (const int*)  d_in[2];
  const int*   img_size = (const int*)  d_in[3];
  const float* W_feat   = (const float*)d_in[4];
  const float* b_feat   = (const float*)d_in[5];
  const float* W_cls    = (const float*)d_in[6];
  const float* b_cls    = (const float*)d_in[7];
  const float* W_score  = (const float*)d_in[8];
  const float* b_score  = (const float*)d_in[9];

  float* pooled = (float*)d_ws;            // 128*1024 floats (512 KB)
  float* fc7    = pooled + (size_t)Rr * Cc; // 128*2048 floats (1 MB)
  float* out    = (float*)d_out;

  // Kernel 1: 128 ROIs x (1024 channels / 256 threads)
  dim3 g1(Rr, Cc / 256);
  roipool_mean_kernel<<<g1, 256, 0, stream>>>(x, rois, roi_idx, img_size, pooled);

  // Kernel 2: 8 M-tiles x 64 N-pairs = 512 waves -> 64 blocks of 8 waves
  gemm_fc7_kernel<<<64, 256, 0, stream>>>(pooled, W_feat, b_feat, fc7);

  // Kernel 3: 8 M-tiles x 7 N-tiles = 56 waves -> 7 blocks of 8 waves
  gemm_heads_kernel<<<7, 256, 0, stream>>>(fc7, W_cls, b_cls, W_score, b_score, out);
}